// FASTopic_48455821034128
// MI455X (gfx1250) — compile-verified
//
#include <hip/hip_runtime.h>
#include <math.h>

#define V_SIZE 50000
#define E_DIM 384
#define K_TOP 100
#define NDOCS 2048
#define N_ITER 100
#define EPS_SINK 1e-16f
#define EPS_LOG 1e-12f
#define TW_NWG 50
#define TW_COLS (V_SIZE / TW_NWG)   // 1000 columns owned per workgroup

typedef __attribute__((ext_vector_type(2))) float v2f;
typedef __attribute__((ext_vector_type(8))) float v8f;

__device__ __forceinline__ v8f wmma_f32_16x16x4(v2f a, v2f b, v8f c) {
  // 8-arg f32 pattern: (neg_a, A, neg_b, B, c_mod, C, reuse_a, reuse_b)
  return __builtin_amdgcn_wmma_f32_16x16x4_f32(false, a, false, b, (short)0, c,
                                               false, false);
}

__device__ __forceinline__ float wave_reduce_sum(float v) {
#pragma unroll
  for (int off = 16; off > 0; off >>= 1) v += __shfl_xor(v, off, 32);
  return v;
}

// ---------------------------------------------------------------- init
__global__ void zero_kernel(float* p, int n) {
  int i = blockIdx.x * blockDim.x + threadIdx.x;
  if (i < n) p[i] = 0.0f;
}

// ------------------------------------------------- row squared norms
__global__ void rowsq_kernel(const float* __restrict__ x,
                             float* __restrict__ out, int rows) {
  int wave = threadIdx.x >> 5;
  int lane = threadIdx.x & 31;
  int row = blockIdx.x * (blockDim.x >> 5) + wave;
  if (row >= rows) return;  // whole wave exits together
  const float* rp = x + (size_t)row * E_DIM;
  float s = 0.f;
#pragma unroll
  for (int i = lane; i < E_DIM; i += 32) { float t = rp[i]; s += t * t; }
  s = wave_reduce_sum(s);
  if (lane == 0) out[row] = s;
}

// ------------------------------------------------- softmax marginals
__device__ void block_softmax(const float* __restrict__ w,
                              float* __restrict__ b, int len, float* red) {
  int tid = threadIdx.x, nt = blockDim.x;
  float mx = -INFINITY;
  for (int i = tid; i < len; i += nt) mx = fmaxf(mx, w[i]);
  red[tid] = mx; __syncthreads();
  for (int s = nt >> 1; s > 0; s >>= 1) {
    if (tid < s) red[tid] = fmaxf(red[tid], red[tid + s]);
    __syncthreads();
  }
  mx = red[0]; __syncthreads();
  float sm = 0.f;
  for (int i = tid; i < len; i += nt) sm += __expf(w[i] - mx);
  red[tid] = sm; __syncthreads();
  for (int s = nt >> 1; s > 0; s >>= 1) {
    if (tid < s) red[tid] += red[tid + s];
    __syncthreads();
  }
  sm = red[0]; __syncthreads();
  float inv = 1.0f / sm;
  for (int i = tid; i < len; i += nt) b[i] = __expf(w[i] - mx) * inv;
}

__global__ void __launch_bounds__(1024)
softmax_prep_kernel(const float* tw, const float* ww, float* bDT, float* bTW) {
  __shared__ float red[1024];
  block_softmax(tw, bDT, K_TOP, red);
  __syncthreads();
  block_softmax(ww, bTW, V_SIZE, red);
}

// --------------------- pairwise sq-euclid + Gibbs kernel, fp32 WMMA
// one wave per 16x16 output tile; K-loop over E in steps of 4.
// Out-of-range rows/cols are CLAMPED (not masked): garbage A-rows /
// B-cols only affect out-of-range outputs which the store guard drops.
// This keeps all loads unconditional -> no EXEC churn, pipelined loads.
__global__ void __launch_bounds__(32)
pairwise_gibbs_kernel(const float* __restrict__ x, const float* __restrict__ y,
                      const float* __restrict__ xsq, const float* __restrict__ ysq,
                      int n, int m, float alpha,
                      float* __restrict__ Mout, float* __restrict__ Kout) {
  int lane = threadIdx.x & 31;
  int hf = lane >> 4;
  int l16 = lane & 15;
  int rowc = min(blockIdx.x * 16 + l16, n - 1);
  int colc = min(blockIdx.y * 16 + l16, m - 1);
  const float* xp = x + (size_t)rowc * E_DIM + 2 * hf;  // even -> 8B aligned
  const float* yp = y + (size_t)colc * E_DIM + 2 * hf;
  v8f acc = {};
  v2f a = *(const v2f*)xp;
  v2f b = *(const v2f*)yp;
#pragma unroll 4
  for (int k = 4; k < E_DIM; k += 4) {   // software-pipelined: fetch k, mma k-4
    v2f an = *(const v2f*)(xp + k);
    v2f bn = *(const v2f*)(yp + k);
    acc = wmma_f32_16x16x4(a, b, acc);
    a = an; b = bn;
  }
  acc = wmma_f32_16x16x4(a, b, acc);
#pragma unroll
  for (int j = 0; j < 8; ++j) {
    int r = blockIdx.x * 16 + j + hf * 8;   // C/D layout: M = j + 8*(lane>=16)
    int c = blockIdx.y * 16 + l16;          // N = lane & 15
    if (r < n && c < m) {
      float Mv = xsq[r] + ysq[c] - 2.0f * acc[j];
      size_t idx = (size_t)r * m + c;
      Mout[idx] = Mv;
      Kout[idx] = __expf(-alpha * Mv);
    }
  }
}

// ---------------------------------------- Sinkhorn DT (single WGP)
__global__ void __launch_bounds__(1024)
sinkhorn_dt_kernel(const float* __restrict__ Kdt, const float* __restrict__ bDT,
                   float* __restrict__ uOut, float* __restrict__ vOut) {
  __shared__ float u[NDOCS];
  __shared__ float v[K_TOP];
  __shared__ float colpart[8][128];
  int tid = threadIdx.x;
  const float a = 1.0f / NDOCS;
  for (int i = tid; i < NDOCS; i += 1024) u[i] = 1.0f / NDOCS;
  __syncthreads();
  int c = tid & 127, g = tid >> 7;
  for (int it = 0; it < N_ITER; ++it) {
    // v[c] = b[c] / (sum_r K[r,c] u[r] + eps)
    float p = 0.f;
    if (c < K_TOP)
      for (int r = g; r < NDOCS; r += 8) p += Kdt[(size_t)r * K_TOP + c] * u[r];
    colpart[g][c] = p;
    __syncthreads();
    if (tid < K_TOP) {
      float s = 0.f;
#pragma unroll
      for (int gg = 0; gg < 8; ++gg) s += colpart[gg][tid];
      v[tid] = bDT[tid] / (s + EPS_SINK);
    }
    __syncthreads();
    // u[r] = a / (sum_c K[r,c] v[c] + eps)
    for (int r = tid; r < NDOCS; r += 1024) {
      float s = 0.f;
      for (int cc = 0; cc < K_TOP; ++cc) s += Kdt[(size_t)r * K_TOP + cc] * v[cc];
      u[r] = a / (s + EPS_SINK);
    }
    __syncthreads();
  }
  for (int i = tid; i < NDOCS; i += 1024) uOut[i] = u[i];
  if (tid < K_TOP) vOut[tid] = v[tid];
}

// --------------- Sinkhorn TW: 50 WGs, ONE global sync per iteration
__global__ void __launch_bounds__(256)
sinkhorn_tw_kernel(const float* __restrict__ Ktw, const float* __restrict__ bTW,
                   float* __restrict__ uOut, float* __restrict__ vOut,
                   float* __restrict__ accRed, int* __restrict__ counters) {
  __shared__ float u[K_TOP];
  __shared__ float vloc[TW_COLS];
  __shared__ float part2[2][128];
  int tid = threadIdx.x;
  int base = blockIdx.x * TW_COLS;
  const float a = 1.0f / K_TOP;
  if (tid < K_TOP) u[tid] = 1.0f / K_TOP;
  __syncthreads();
  int kk = tid & 127, g = tid >> 7;
  for (int it = 0; it < N_ITER; ++it) {
    // v-update for owned columns (lane-coalesced over w)
    for (int wi = tid; wi < TW_COLS; wi += 256) {
      size_t w = base + wi;
      float s = 0.f;
      for (int k = 0; k < K_TOP; ++k) s += Ktw[(size_t)k * V_SIZE + w] * u[k];
      vloc[wi] = bTW[w] / (s + EPS_SINK);
    }
    __syncthreads();
    // partial Kv over owned columns
    float p = 0.f;
    if (kk < K_TOP) {
      const float* krow = Ktw + (size_t)kk * V_SIZE + base + g * (TW_COLS / 2);
      const float* vp = vloc + g * (TW_COLS / 2);
      for (int wi = 0; wi < TW_COLS / 2; ++wi) p += krow[wi] * vp[wi];
    }
    part2[g][kk] = p;
    __syncthreads();
    if (tid < K_TOP)
      atomicAdd(&accRed[(size_t)it * K_TOP + tid], part2[0][tid] + part2[1][tid]);
    __syncthreads();
    if (tid == 0) {  // single global barrier per iteration
      __threadfence();
      atomicAdd(&counters[it], 1);
      while (__hip_atomic_load(&counters[it], __ATOMIC_ACQUIRE,
                               __HIP_MEMORY_SCOPE_AGENT) < TW_NWG)
        __builtin_amdgcn_s_sleep(1);
    }
    __syncthreads();
    if (tid < K_TOP) {  // every WG computes u locally (no 2nd barrier)
      float tot = __hip_atomic_load(&accRed[(size_t)it * K_TOP + tid],
                                    __ATOMIC_RELAXED, __HIP_MEMORY_SCOPE_AGENT);
      u[tid] = a / (tot + EPS_SINK);
    }
    __syncthreads();
  }
  for (int wi = tid; wi < TW_COLS; wi += 256) vOut[base + wi] = vloc[wi];
  if (blockIdx.x == 0 && tid < K_TOP) uOut[tid] = u[tid];
}

// ---------------------------------------------------------------- svec
__global__ void svec_kernel(const float* vDT, const float* uTW, float* s) {
  int i = threadIdx.x;
  if (i < K_TOP) s[i] = vDT[i] * uTW[i];
}

// ------------------------------------------------- loss_ETP reduction
__global__ void etp_loss_kernel(const float* __restrict__ Kmat,
                                const float* __restrict__ Mmat,
                                const float* __restrict__ u,
                                const float* __restrict__ v,
                                int n, int m, float* __restrict__ acc) {
  size_t total = (size_t)n * m;
  float local = 0.f;
  for (size_t i = blockIdx.x * (size_t)blockDim.x + threadIdx.x; i < total;
       i += (size_t)gridDim.x * blockDim.x) {
    int r = (int)(i / m), c = (int)(i % m);
    local += u[r] * Kmat[i] * v[c] * Mmat[i];
  }
  local = wave_reduce_sum(local);
  if ((threadIdx.x & 31) == 0) atomicAdd(acc, local);
}

// --------- fused recon GEMM + log + bow-dot (never materializes recon)
__global__ void __launch_bounds__(128)
recon_dsr_kernel(const float* __restrict__ bow, const float* __restrict__ Kdt,
                 const float* __restrict__ Ktw, const float* __restrict__ uDT,
                 const float* __restrict__ vTW, const float* __restrict__ svec,
                 float* __restrict__ acc) {
  int lane = threadIdx.x & 31;
  int wave = threadIdx.x >> 5;
  int hf = lane >> 4, l16 = lane & 15;
  int doc0 = (blockIdx.x * 4 + wave) * 16;  // 128 doc tiles exact
  int col = blockIdx.y * 16 + l16;          // 3125 word tiles exact
#pragma unroll
  for (int j = 0; j < 8; ++j) {             // warm bow tile while WMMA runs
    int doc = doc0 + j + hf * 8;
    __builtin_prefetch(bow + (size_t)doc * V_SIZE + col, 0, 1);
  }
  int docA = doc0 + l16;
  const float* ap = Kdt + (size_t)docA * K_TOP;
  v8f accv = {};
  int ka0 = 2 * hf;
  v2f a = *(const v2f*)(ap + ka0);           // even offset -> aligned
  a.x *= svec[ka0];
  a.y *= svec[ka0 + 1];
  v2f b;
  b.x = Ktw[(size_t)ka0 * V_SIZE + col];
  b.y = Ktw[(size_t)(ka0 + 1) * V_SIZE + col];
#pragma unroll 4
  for (int k = 4; k < K_TOP; k += 4) {       // pipelined: fetch k, mma k-4
    int ka = k + 2 * hf;
    v2f an = *(const v2f*)(ap + ka);
    an.x *= svec[ka];
    an.y *= svec[ka + 1];
    v2f bn;
    bn.x = Ktw[(size_t)ka * V_SIZE + col];
    bn.y = Ktw[(size_t)(ka + 1) * V_SIZE + col];
    accv = wmma_f32_16x16x4(a, b, accv);
    a = an; b = bn;
  }
  accv = wmma_f32_16x16x4(a, b, accv);
  float cs = (float)K_TOP * vTW[col];
  float partial = 0.f;
#pragma unroll
  for (int j = 0; j < 8; ++j) {
    int doc = doc0 + j + hf * 8;
    float rs = (float)NDOCS * uDT[doc];
    float recon = rs * cs * accv[j];
    partial += bow[(size_t)doc * V_SIZE + col] * __logf(recon + EPS_LOG);
  }
  partial = wave_reduce_sum(partial);
  if (lane == 0) atomicAdd(acc, partial);
}

__global__ void finalize_kernel(const float* accETP, const float* accDSR,
                                float* out) {
  out[0] = accETP[0] - accDSR[0] / (float)NDOCS;
}

extern "C" void kernel_launch(void* const* d_in, const int* in_sizes, int n_in,
                              void* d_out, int out_size, void* d_ws,
                              size_t ws_size, hipStream_t stream) {
  (void)in_sizes; (void)n_in; (void)out_size; (void)ws_size;
  const float* bow = (const float*)d_in[0];
  const float* docE = (const float*)d_in[1];
  const float* wordE = (const float*)d_in[2];
  const float* topE = (const float*)d_in[3];
  const float* tw = (const float*)d_in[4];
  const float* ww = (const float*)d_in[5];
  // d_in[6] (indices) unused: is_OT=False path
  float* out = (float*)d_out;

  float* ws = (float*)d_ws;
  size_t off = 0;
  auto alloc = [&](size_t n) { float* p = ws + off; off += n; return p; };
  float* Mdt = alloc((size_t)NDOCS * K_TOP);
  float* Kdt = alloc((size_t)NDOCS * K_TOP);
  float* Mtw = alloc((size_t)K_TOP * V_SIZE);
  float* Ktw = alloc((size_t)K_TOP * V_SIZE);
  float* docSq = alloc(NDOCS);
  float* topSq = alloc(K_TOP);
  float* wordSq = alloc(V_SIZE);
  float* bDT = alloc(K_TOP);
  float* bTW = alloc(V_SIZE);
  float* uDT = alloc(NDOCS);
  float* vDT = alloc(K_TOP);
  float* uTW = alloc(K_TOP);
  float* vTW = alloc(V_SIZE);
  float* svec = alloc(K_TOP);
  float* zbase = ws + off;                  // zero-init region start
  float* accETP = alloc(1);
  float* accDSR = alloc(1);
  float* accRed = alloc((size_t)N_ITER * K_TOP);
  int* counters = (int*)alloc(N_ITER);
  int zcount = 2 + N_ITER * K_TOP + N_ITER;

  zero_kernel<<<(zcount + 255) / 256, 256, 0, stream>>>(zbase, zcount);
  rowsq_kernel<<<(NDOCS + 7) / 8, 256, 0, stream>>>(docE, docSq, NDOCS);
  rowsq_kernel<<<(K_TOP + 7) / 8, 256, 0, stream>>>(topE, topSq, K_TOP);
  rowsq_kernel<<<(V_SIZE + 7) / 8, 256, 0, stream>>>(wordE, wordSq, V_SIZE);
  softmax_prep_kernel<<<1, 1024, 0, stream>>>(tw, ww, bDT, bTW);
  dim3 gDT((NDOCS + 15) / 16, (K_TOP + 15) / 16);
  pairwise_gibbs_kernel<<<gDT, 32, 0, stream>>>(docE, topE, docSq, topSq,
                                                NDOCS, K_TOP, 3.0f, Mdt, Kdt);
  dim3 gTW((K_TOP + 15) / 16, (V_SIZE + 15) / 16);
  pairwise_gibbs_kernel<<<gTW, 32, 0, stream>>>(topE, wordE, topSq, wordSq,
                                                K_TOP, V_SIZE, 2.0f, Mtw, Ktw);
  sinkhorn_dt_kernel<<<1, 1024, 0, stream>>>(Kdt, bDT, uDT, vDT);
  sinkhorn_tw_kernel<<<TW_NWG, 256, 0, stream>>>(Ktw, bTW, uTW, vTW, accRed,
                                                 counters);
  svec_kernel<<<1, 128, 0, stream>>>(vDT, uTW, svec);
  etp_loss_kernel<<<1024, 256, 0, stream>>>(Kdt, Mdt, uDT, vDT, NDOCS, K_TOP,
                                            accETP);
  etp_loss_kernel<<<4096, 256, 0, stream>>>(Ktw, Mtw, uTW, vTW, K_TOP, V_SIZE,
                                            accETP);
  dim3 gR(32, V_SIZE / 16);
  recon_dsr_kernel<<<gR, 128, 0, stream>>>(bow, Kdt, Ktw, uDT, vTW, svec,
                                           accDSR);
  finalize_kernel<<<1, 1, 0, stream>>>(accETP, accDSR, out);
}